// RootsToPolynomial_88046829568564
// MI455X (gfx1250) — compile-verified
//
#include <hip/hip_runtime.h>

// CDNA5 / gfx1250: roots -> monic polynomial coefficients via a product tree.
// Big levels run as fp32 WMMA (V_WMMA_F32_16X16X4_F32) over Toeplitz tiles,
// chained along equal-output diagonals in the D accumulator; zero-padded LDS
// slots make all fragment loads and ds_add_f32 scatters unconditional.

typedef float v2f __attribute__((ext_vector_type(2)));
typedef float v8f __attribute__((ext_vector_type(8)));

#define ORDER    1024
#define OUT_LEN  (ORDER + 1)
#define NTHREADS 256
#define NWAVES   (NTHREADS / 32)
#define FRONT    4              // zeroed floats before each poly (u[i-4] min)
#define PADTOT   80             // slot = Lout + 80 (front 4 + back 76)
#define BUFSZ    2336           // >= max(2048 level0, 16*(65+80)=2320)

__global__ __launch_bounds__(NTHREADS)
void RootsToPolynomial_88046829568564_kernel(const float* __restrict__ x,
                                             float* __restrict__ out)
{
    __shared__ float bufA[BUFSZ];
    __shared__ float bufB[BUFSZ];

    const int row  = blockIdx.x;
    const int tid  = threadIdx.x;
    const int wave = tid >> 5;
    const int lane = tid & 31;
    const int m    = lane & 15;     // A row / B col / D col
    const int hi   = lane >> 4;     // lane-half select
    const int p0   = hi << 1;       // K-dim base: 0 or 2

    const float* __restrict__ xrow = x + (size_t)row * ORDER;

    // ---- Level 0: 1024 linear factors [1, -x_i] (packed layout) ----
    for (int i = tid; i < ORDER; i += NTHREADS) {
        bufA[2 * i]     = 1.0f;
        bufA[2 * i + 1] = -xrow[i];
    }
    __syncthreads();

    float* src = bufA;
    float* dst = bufB;
    int P = ORDER, L = 2;
    int slotIn = 2, offIn = 0;      // per-poly stride / data offset in src

    while (P > 1) {
        const int  pairs   = P >> 1;
        const int  Lout    = 2 * L - 1;
        const bool padOut  = (Lout >= 65);           // feeds a WMMA level
        const int  slotOut = padOut ? (Lout + PADTOT) : Lout;
        const int  offOut  = padOut ? FRONT : 0;

        if (L < 65) {
            // -------- small levels: plain VALU f32 FMA conv --------
            if (padOut) {   // zero pad regions once (only the Lout=65 level)
                for (int idx = tid; idx < pairs * slotOut; idx += NTHREADS)
                    dst[idx] = 0.0f;
                __syncthreads();
            }
            const int total = pairs * Lout;
            for (int idx = tid; idx < total; idx += NTHREADS) {
                const int q = idx / Lout;
                const int k = idx - q * Lout;
                const float* u = src + (2 * q) * slotIn + offIn;
                const float* v = src + (2 * q + 1) * slotIn + offIn;
                int ilo = k - (L - 1); if (ilo < 0) ilo = 0;
                int ihi = k;           if (ihi > L - 1) ihi = L - 1;
                float acc = 0.0f;
                for (int i = ilo; i <= ihi; ++i)
                    acc = fmaf(u[i], v[k - i], acc);
                dst[q * slotOut + offOut + k] = acc;
            }
            __syncthreads();
        } else {
            // -------- big levels: fp32 WMMA Toeplitz tiles --------
            for (int idx = tid; idx < pairs * slotOut; idx += NTHREADS)
                dst[idx] = 0.0f;                     // also zeroes the pads
            __syncthreads();

            const int nTilesU = (L + 2) / 16 + 1;
            const int nTilesV = (L - 1) / 64 + 1;
            const int nS      = nTilesU + 4 * nTilesV - 4;  // diagonals s=a+4b
            const int nTasks  = pairs * nS;

            // wave-uniform task loop: EXEC all-ones at the WMMA
            for (int task = wave; task < nTasks; task += NWAVES) {
                const int q = task / nS;
                const int s = task - q * nS;
                const float* u = src + (2 * q) * slotIn + offIn;
                const float* v = src + (2 * q + 1) * slotIn + offIn;

                // b range so that a = s-4b lies in [0, nTilesU)
                int bLo = s - nTilesU + 4;
                bLo = (bLo > 0) ? (bLo >> 2) : 0;
                const int bHi0 = s >> 2;
                const int bHi  = (bHi0 < nTilesV - 1) ? bHi0 : (nTilesV - 1);

                // chain all tiles of this diagonal in the D accumulator
                v8f Cf = {};
                for (int b = bLo; b <= bHi; ++b) {
                    const int c1 = 16 * (s - 4 * b);
                    const int c2 = 64 * b;
                    const int i0 = c1 + m - p0;        // >= -3, < L+18 (padded)
                    const int j0 = c2 + 4 * m + p0;    // < L+63      (padded)
                    v2f Af, Bf;
                    Af.x = u[i0];
                    Af.y = u[i0 - 1];
                    Bf.x = v[j0];
                    Bf.y = v[j0 + 1];
                    // D[m][n] += sum_p u[c1+m-p] * v[c2+4n+p]
                    Cf = __builtin_amdgcn_wmma_f32_16x16x4_f32(
                            false, Af, false, Bf, (short)0, Cf, false, false);
                }

                // one scatter per diagonal: k = 16s + M + 4n, M = r + 8*hi
                float* c = dst + q * slotOut + offOut + 16 * s + 8 * hi + 4 * m;
                #pragma unroll
                for (int r = 0; r < 8; ++r)
                    atomicAdd(&c[r], Cf[r]);           // ds_add_f32, no guard
            }
            __syncthreads();
        }

        float* t = src; src = dst; dst = t;
        P = pairs; L = Lout;
        slotIn = slotOut; offIn = offOut;
    }

    // src holds the final length-1025 coefficient vector at offset offIn
    float* __restrict__ orow = out + (size_t)row * OUT_LEN;
    for (int k = tid; k < OUT_LEN; k += NTHREADS)
        orow[k] = src[offIn + k];
}

extern "C" void kernel_launch(void* const* d_in, const int* in_sizes, int n_in,
                              void* d_out, int out_size, void* d_ws, size_t ws_size,
                              hipStream_t stream)
{
    (void)n_in; (void)out_size; (void)d_ws; (void)ws_size;
    const float* x = (const float*)d_in[0];
    float* out = (float*)d_out;
    const int batch = in_sizes[0] / ORDER;   // 8192
    RootsToPolynomial_88046829568564_kernel<<<batch, NTHREADS, 0, stream>>>(x, out);
}